// MacroNotchOp_28647431864381
// MI455X (gfx1250) — compile-verified
//
#include <hip/hip_runtime.h>

// MacroNotchOp: sum over i<j of relu(2 - (relu(|xi-xj| - (sxi+sxj)/2) + relu(|yi-yj| - (syi+syj)/2)))^2
// N = 4096 movable, pos has x in [0,4096), y in [4096, 8192). macro_mask is unused by the reference.

#define NMOV   4096
#define NPHYS  4096
#define TILE   128
#define TGRID  (NMOV / TILE)      // 32  -> 32x32 block grid, upper triangle active
#define NOTCH  2.0f

typedef float v2f __attribute__((ext_vector_type(2)));
typedef float v8f __attribute__((ext_vector_type(8)));

// Exact f32 wave32 reduction via V_WMMA_F32_16X16X4_F32 with an all-ones B matrix.
// A layout (16x4 f32): lane l<16 holds (M=l,K=0),(M=l,K=1); lane l>=16 holds (M=l-16,K=2),(M=l-16,K=3).
// With a = {acc, 0}: D[m,n] = acc_m + acc_{m+16} (same for every n).
// C/D layout: VGPR r holds M=r (lanes 0-15) / M=r+8 (lanes 16-31). So per-lane sum of the 8 D
// registers gives half the total; shfl_xor(16) combines the halves. Fixed HW order => deterministic.
__device__ __forceinline__ float wave_reduce_f32(float acc) {
#if __has_builtin(__builtin_amdgcn_wmma_f32_16x16x4_f32)
    v2f a = { acc, 0.0f };
    v2f b = { 1.0f, 1.0f };
    v8f c = {};
    c = __builtin_amdgcn_wmma_f32_16x16x4_f32(false, a, false, b, (short)0, c, false, false);
    float s = ((c[0] + c[1]) + (c[2] + c[3])) + ((c[4] + c[5]) + (c[6] + c[7]));
    s += __shfl_xor(s, 16, 32);
    return s;
#else
    for (int m = 16; m > 0; m >>= 1) acc += __shfl_xor(acc, m, 32);
    return acc;
#endif
}

__global__ __launch_bounds__(TILE) void notch_pair_kernel(
    const float* __restrict__ pos,
    const float* __restrict__ sx,
    const float* __restrict__ sy,
    float* __restrict__ partials) {
    __shared__ float lx[TILE], ly[TILE], lhx[TILE], lhy[TILE];
    __shared__ float wsum[TILE / 32];

    const int bi = blockIdx.x;          // i-tile
    const int bj = blockIdx.y;          // j-tile
    const int t  = threadIdx.x;
    const int pidx = bi * TGRID + bj;

    if (bj < bi) {                      // lower triangle: contribute exactly 0
        if (t == 0) partials[pidx] = 0.0f;
        return;
    }

    // Register-resident i data (one i per thread)
    const int   i   = bi * TILE + t;
    const float xi  = pos[i];
    const float yi  = pos[NPHYS + i];
    const float hxi = 0.5f * sx[i];
    const float hyi = 0.5f * sy[i];

    // Stage the j-tile through LDS (pre-halved sizes to save a VALU op per pair)
    const int j0 = bj * TILE;
    lx[t]  = pos[j0 + t];
    ly[t]  = pos[NPHYS + j0 + t];
    lhx[t] = 0.5f * sx[j0 + t];
    lhy[t] = 0.5f * sy[j0 + t];
    __syncthreads();

    float acc = 0.0f;
    if (bi == bj) {
        // Diagonal tile: need the j > i predicate
        #pragma unroll 4
        for (int jj = 0; jj < TILE; ++jj) {
            float gx = fmaxf(fabsf(xi - lx[jj]) - (hxi + lhx[jj]), 0.0f);
            float gy = fmaxf(fabsf(yi - ly[jj]) - (hyi + lhy[jj]), 0.0f);
            float p  = fmaxf(NOTCH - (gx + gy), 0.0f);
            float p2 = p * p;
            acc += (j0 + jj > i) ? p2 : 0.0f;
        }
    } else {
        // Off-diagonal tile: every pair has j > i, branch-free inner loop
        #pragma unroll 8
        for (int jj = 0; jj < TILE; ++jj) {
            float gx = fmaxf(fabsf(xi - lx[jj]) - (hxi + lhx[jj]), 0.0f);
            float gy = fmaxf(fabsf(yi - ly[jj]) - (hyi + lhy[jj]), 0.0f);
            float p  = fmaxf(NOTCH - (gx + gy), 0.0f);
            acc += p * p;
        }
    }

    // Wave reduction (WMMA), then fixed-order combine of the 4 waves.
    // All TILE threads are active here -> EXEC all ones as WMMA requires.
    float s = wave_reduce_f32(acc);
    if ((t & 31) == 0) wsum[t >> 5] = s;
    __syncthreads();
    if (t == 0) {
        float tot = 0.0f;
        #pragma unroll
        for (int w = 0; w < TILE / 32; ++w) tot += wsum[w];
        partials[pidx] = tot;
    }
}

__global__ __launch_bounds__(256) void notch_final_kernel(
    const float* __restrict__ partials,   // TGRID*TGRID = 1024 values
    float* __restrict__ out) {
    __shared__ float wsum[8];
    const int t = threadIdx.x;

    float acc = 0.0f;
    #pragma unroll
    for (int k = 0; k < (TGRID * TGRID) / 256; ++k)
        acc += partials[t + k * 256];

    float s = wave_reduce_f32(acc);       // 256 threads all active
    if ((t & 31) == 0) wsum[t >> 5] = s;
    __syncthreads();
    if (t == 0) {
        float tot = 0.0f;
        #pragma unroll
        for (int w = 0; w < 8; ++w) tot += wsum[w];
        out[0] = tot;
    }
}

extern "C" void kernel_launch(void* const* d_in, const int* in_sizes, int n_in,
                              void* d_out, int out_size, void* d_ws, size_t ws_size,
                              hipStream_t stream) {
    const float* pos = (const float*)d_in[0];   // 8192 floats: x then y
    // d_in[1] = macro_mask (unused by reference)
    const float* sx  = (const float*)d_in[2];   // 4096 floats
    const float* sy  = (const float*)d_in[3];   // 4096 floats

    float* partials = (float*)d_ws;             // needs TGRID*TGRID*4 = 4 KB scratch

    dim3 grid(TGRID, TGRID);
    notch_pair_kernel<<<grid, TILE, 0, stream>>>(pos, sx, sy, partials);
    notch_final_kernel<<<1, 256, 0, stream>>>(partials, (float*)d_out);
}